// InfoNCELoss_36352603193982
// MI455X (gfx1250) — compile-verified
//
#include <hip/hip_runtime.h>

typedef float v2f __attribute__((ext_vector_type(2)));
typedef float v8f __attribute__((ext_vector_type(8)));

#define DIM 64
#define TAU_INV 20.0f
#define LOG2E 1.44269504f
#define LN2 0.69314718f
#define A_SCALE (TAU_INV * LOG2E)     // fold tau and log2(e) into z1 rows

#define WAVES 8
#define ROWS_PER_WAVE 32              // two 16-row WMMA subtiles per wave
#define ROWS_PER_BLOCK (WAVES * ROWS_PER_WAVE)   // 256
#define COL_SPLIT 2                   // split column sweep across 2 blocks
#define LDS_STRIDE 68                 // 64 + 4 pad: 17 x 16B, coprime with 64 banks
#define TILE_F (16 * LDS_STRIDE)      // floats per LDS B-tile buffer

// ---------------------------------------------------------------------------
// CDNA5 async global->LDS copy (ASYNCcnt), per CDNA5 ISA
// ---------------------------------------------------------------------------
__device__ __forceinline__ void async_load_b128(unsigned lds_byte_addr,
                                                unsigned long long gaddr) {
    asm volatile("global_load_async_to_lds_b128 %0, %1, off"
                 :: "v"(lds_byte_addr), "v"(gaddr)
                 : "memory");
}
__device__ __forceinline__ void wait_async_le1() {
    asm volatile("s_wait_asynccnt 0x1" ::: "memory");
}
__device__ __forceinline__ void wait_async_0() {
    asm volatile("s_wait_asynccnt 0x0" ::: "memory");
}

// ---------------------------------------------------------------------------
// Kernel 1: row L2-normalize (one wave32 per row, 2 floats/lane).
// z1 rows additionally pre-scaled by TAU_INV*log2(e) so logits are base-2.
// ---------------------------------------------------------------------------
__global__ __launch_bounds__(256) void infonce_normalize(
    const float* __restrict__ z1, const float* __restrict__ z2,
    float* __restrict__ z1s, float* __restrict__ z2n, int n)
{
    const int gw   = (blockIdx.x * blockDim.x + threadIdx.x) >> 5; // row id
    const int lane = threadIdx.x & 31;
    const bool first = (gw < n);
    const float* src = first ? z1 : z2;
    float*       dst = first ? z1s : z2n;
    const int row = first ? gw : gw - n;
    const float scale = first ? A_SCALE : 1.0f;

    float2 v = ((const float2*)(src + (size_t)row * DIM))[lane];
    float ss = v.x * v.x + v.y * v.y;
    #pragma unroll
    for (int off = 16; off; off >>= 1) ss += __shfl_xor(ss, off);
    const float inv = scale / fmaxf(sqrtf(ss), 1e-12f);
    float2 o; o.x = v.x * inv; o.y = v.y * inv;
    ((float2*)(dst + (size_t)row * DIM))[lane] = o;
}

// ---------------------------------------------------------------------------
// Kernel 2: fused sim GEMM (f32 WMMA) + streaming row sumexp + diag capture.
// Each wave owns 32 rows (2 subtiles sharing B); block sweeps half the cols
// through a double-buffered async-staged LDS B-tile.
// ---------------------------------------------------------------------------
__global__ __launch_bounds__(256) void infonce_main(
    const float* __restrict__ z1s, const float* __restrict__ z2n,
    float* __restrict__ dsumg, float* __restrict__ accum, int n)
{
    __shared__ float bt[2 * TILE_F];

    const int tid  = threadIdx.x;
    const int lane = tid & 31;
    const int wave = tid >> 5;
    const int half = lane >> 4;
    const int g    = lane & 15;

    const int rowgroup = blockIdx.x >> 1;
    const int chalf    = blockIdx.x & 1;
    const int r0 = rowgroup * ROWS_PER_BLOCK + wave * ROWS_PER_WAVE;

    const int ntiles = n >> 4;
    const int htiles = ntiles / COL_SPLIT;
    const int ct0    = chalf * htiles;

    // Staging slice for this thread: 16B of one B-tile row.
    const int scol = tid >> 4;             // 0..15 : z2 row within tile
    const int sk   = (tid & 15) * 4;       // 0..60 : K offset
    const unsigned bt_base = (unsigned)(unsigned long long)(const void*)bt; // LDS byte offset
    const unsigned stage_lds = bt_base + (unsigned)(scol * LDS_STRIDE + sk) * 4u;

    // A tiles: 2 x (16 rows x 64 K) in WMMA f32 16x16x4 layout (ISA 7.12.2).
    v2f a0[16], a1[16];
    {
        const float* ar0 = z1s + (size_t)(r0 + g) * DIM + 2 * half;
        const float* ar1 = z1s + (size_t)(r0 + 16 + g) * DIM + 2 * half;
        #pragma unroll
        for (int kt = 0; kt < 16; ++kt) {
            float2 t0 = *(const float2*)(ar0 + 4 * kt);
            float2 t1 = *(const float2*)(ar1 + 4 * kt);
            a0[kt].x = t0.x; a0[kt].y = t0.y;
            a1[kt].x = t1.x; a1[kt].y = t1.y;
        }
    }

    float ds0[8], ds1[8];
    #pragma unroll
    for (int v = 0; v < 8; ++v) { ds0[v] = 0.0f; ds1[v] = 0.0f; }
    float possum = 0.0f;
    const int dt0 = r0 >> 4, dt1 = dt0 + 1;

    // Prologue: stage first tile into buffer 0.
    {
        unsigned long long gp = (unsigned long long)(const void*)
            (z2n + (size_t)(ct0 * 16 + scol) * DIM + sk);
        async_load_b128(stage_lds, gp);
    }

    for (int i = 0; i < htiles; ++i) {
        const int ct = ct0 + i;
        if (i + 1 < htiles) {   // stage next tile into the other buffer
            unsigned long long gp = (unsigned long long)(const void*)
                (z2n + (size_t)((ct + 1) * 16 + scol) * DIM + sk);
            async_load_b128(stage_lds + (unsigned)(((i + 1) & 1) * TILE_F) * 4u, gp);
            wait_async_le1();   // tile ct complete (async loads retire in order)
        } else {
            wait_async_0();
        }
        __syncthreads();        // tile ct visible from every wave's copies

        const float* bp = bt + (i & 1) * TILE_F + g * LDS_STRIDE + 2 * half;
        v8f acc0, acc1;
        #pragma unroll
        for (int v = 0; v < 8; ++v) { acc0[v] = 0.0f; acc1[v] = 0.0f; }

        #pragma unroll
        for (int kt = 0; kt < 16; ++kt) {
            const float2 t = *(const float2*)(bp + 4 * kt);
            v2f b; b.x = t.x; b.y = t.y;
            acc0 = __builtin_amdgcn_wmma_f32_16x16x4_f32(
                false, a0[kt], false, b, (short)0, acc0, false, false);
            acc1 = __builtin_amdgcn_wmma_f32_16x16x4_f32(
                false, a1[kt], false, b, (short)0, acc1, false, false);
        }

        // Diagonal capture (uniform branch, taken <=2 of 512 iterations).
        if (ct == dt0) {
            #pragma unroll
            for (int v = 0; v < 8; ++v)
                if (g == v + 8 * half) possum += acc0[v] * LN2;
        }
        if (ct == dt1) {
            #pragma unroll
            for (int v = 0; v < 8; ++v)
                if (g == v + 8 * half) possum += acc1[v] * LN2;
        }

        // Bounded base-2 logits: max-free streaming sumexp, bare v_exp_f32.
        #pragma unroll
        for (int v = 0; v < 8; ++v) {
            ds0[v] += __builtin_amdgcn_exp2f(acc0[v]);
            ds1[v] += __builtin_amdgcn_exp2f(acc1[v]);
        }
        __syncthreads();        // everyone done reading buf before re-stage
    }

    // Per-row partial sums: reduce across the 16-lane group, publish to ws.
    #pragma unroll
    for (int v = 0; v < 8; ++v) {
        float d0 = ds0[v], d1 = ds1[v];
        #pragma unroll
        for (int off = 8; off; off >>= 1) {
            d0 += __shfl_xor(d0, off);
            d1 += __shfl_xor(d1, off);
        }
        if (g == v) {   // lanes v and v+16 publish their half's rows
            atomicAdd(&dsumg[r0 + v + 8 * half], d0);
            atomicAdd(&dsumg[r0 + 16 + v + 8 * half], d1);
        }
    }
    #pragma unroll
    for (int off = 16; off; off >>= 1) possum += __shfl_xor(possum, off);
    if (lane == 0) atomicAdd(&accum[1], possum);
}

// ---------------------------------------------------------------------------
__global__ void infonce_zero(float* __restrict__ dsumg, float* __restrict__ accum, int n)
{
    const int i = blockIdx.x * blockDim.x + threadIdx.x;
    if (i < n) dsumg[i] = 0.0f;
    if (i < 2) accum[i] = 0.0f;
}

// Kernel 3: per-row lse = ln(sumexp), block-reduced into accum[0].
__global__ __launch_bounds__(256) void infonce_rowlse(
    const float* __restrict__ dsumg, float* __restrict__ accum)
{
    __shared__ float partial[WAVES];
    const int i = blockIdx.x * 256 + threadIdx.x;
    float t = __builtin_amdgcn_logf(dsumg[i]) * LN2;   // v_log_f32 is log2
    #pragma unroll
    for (int off = 16; off; off >>= 1) t += __shfl_xor(t, off);
    const int lane = threadIdx.x & 31, wave = threadIdx.x >> 5;
    if (lane == 0) partial[wave] = t;
    __syncthreads();
    if (threadIdx.x == 0) {
        float s = 0.0f;
        #pragma unroll
        for (int w = 0; w < WAVES; ++w) s += partial[w];
        atomicAdd(&accum[0], s);
    }
}

__global__ void infonce_finalize(const float* __restrict__ accum,
                                 float* __restrict__ out, float invn)
{
    out[0] = (accum[0] - accum[1]) * invn;
}

// ---------------------------------------------------------------------------
extern "C" void kernel_launch(void* const* d_in, const int* in_sizes, int n_in,
                              void* d_out, int out_size, void* d_ws, size_t ws_size,
                              hipStream_t stream)
{
    const float* z1 = (const float*)d_in[0];
    const float* z2 = (const float*)d_in[1];
    const int n = in_sizes[0] / DIM;            // 16384

    float* z1s   = (float*)d_ws;                // tau/log2e pre-scaled z1n
    float* z2n   = z1s + (size_t)n * DIM;
    float* dsumg = z2n + (size_t)n * DIM;       // per-row sumexp accumulators
    float* accum = dsumg + n;                   // [0]=sum(lse), [1]=sum(pos)

    infonce_zero<<<n / 256, 256, 0, stream>>>(dsumg, accum, n);

    infonce_normalize<<<(2 * n) / 8, 256, 0, stream>>>(z1, z2, z1s, z2n, n);

    infonce_main<<<(n / ROWS_PER_BLOCK) * COL_SPLIT, 256, 0, stream>>>(
        z1s, z2n, dsumg, accum, n);

    infonce_rowlse<<<n / 256, 256, 0, stream>>>(dsumg, accum);

    infonce_finalize<<<1, 1, 0, stream>>>(accum, (float*)d_out, 1.0f / (float)n);
}